// CausalSelfAttention_89902255440472
// MI455X (gfx1250) — compile-verified
//
#include <hip/hip_runtime.h>
#include <hip/hip_bf16.h>
#include <math.h>

// ---------------------------------------------------------------------------
// Causal self-attention forward for gfx1250 (MI455X), full fp32 via
// V_WMMA_F32_16X16X4_F32, with CDNA5 async global->LDS double-buffered GEMMs.
// B=4, S=2048, D_MODEL=1024, H=16, HD=64.
// ---------------------------------------------------------------------------

typedef __attribute__((ext_vector_type(2))) float v2f;
typedef __attribute__((ext_vector_type(4))) float v4f;
typedef __attribute__((ext_vector_type(8))) float v8f;

#define WMMA4(a, b, c) \
    __builtin_amdgcn_wmma_f32_16x16x4_f32(false, (a), false, (b), (short)0, (c), false, false)

#define B_SZ     4
#define SEQ      2048
#define DMODEL   1024
#define NHEAD    16
#define HDIM     64
#define MROWS    (B_SZ * SEQ)          // 8192
#define LOG10K   9.210340371976184f    // ln(10000)

// ---- CDNA5 async tensor path (inline asm; ASYNCcnt-tracked, bypasses VGPRs)
__device__ __forceinline__ void async_ld16(const void* gmem, void* lds)
{
    // LDS byte address (addrspace(3) offset) in a VGPR, 64-bit global addr pair.
    unsigned loff = (unsigned)(unsigned long long)
        ((__attribute__((address_space(3))) char*)lds);
    unsigned long long ga = (unsigned long long)gmem;
    asm volatile("global_load_async_to_lds_b128 %0, %1, off"
                 :: "v"(loff), "v"(ga) : "memory");
}
__device__ __forceinline__ void wait_async0()
{
    asm volatile("s_wait_asynccnt 0x0" ::: "memory");
}

// ===========================================================================
// Kernel 1: qkv = x @ W_qkv, fused RoPE, scatter to Q/K/V [B,H,S,64]
// Block: 256 threads (8 waves). Tile 64(M) x 128(N), BK=16, double-buffered
// async global->LDS staging. Wave w owns rows (w&3)*16, cols (w>>2)*64.
// ===========================================================================
__global__ __launch_bounds__(256) void qkv_rope_kernel(
    const float* __restrict__ X,   // [8192, 1024]
    const float* __restrict__ W,   // [1024, 3072]
    float* __restrict__ Q,         // [B,H,S,64]
    float* __restrict__ K,
    float* __restrict__ V)
{
    const int N = 3 * DMODEL;                    // 3072
    __shared__ __align__(16) float As[2][64][16];
    __shared__ __align__(16) float Bs[2][16][128];

    const int tid  = threadIdx.x;
    const int lane = tid & 31;
    const int hl   = lane >> 4;                  // half-wave (0/1)
    const int ln   = lane & 15;
    const int wave = tid >> 5;
    const int wm   = (wave & 3) * 16;            // wave row offset in tile
    const int wn   = (wave >> 2) * 64;           // wave col offset in tile
    const int m0   = blockIdx.x * 64;
    const int n0   = blockIdx.y * 128;

    // per-thread staging coordinates
    const int ar = tid >> 2, ac = (tid & 3) * 4;         // A: 64x16, 1 x b128
    const int br0 = tid >> 5,        bc0 = (tid & 31) * 4;   // B: 16x128, 2 x b128
    const int br1 = (tid + 256) >> 5, bc1 = bc0;

    v8f acc[4] = {};

    // prologue: async-stage tile 0 into buffer 0
    {
        const int k0 = 0;
        async_ld16(&X[(size_t)(m0 + ar) * DMODEL + k0 + ac], &As[0][ar][ac]);
        async_ld16(&W[(size_t)(k0 + br0) * N + n0 + bc0],   &Bs[0][br0][bc0]);
        async_ld16(&W[(size_t)(k0 + br1) * N + n0 + bc1],   &Bs[0][br1][bc1]);
    }

    for (int kt = 0; kt < DMODEL / 16; ++kt) {
        const int p = kt & 1;
        wait_async0();
        __syncthreads();                         // tile kt resident in buf p

        if (kt + 1 < DMODEL / 16) {              // async-stage tile kt+1
            const int k0 = (kt + 1) * 16;
            async_ld16(&X[(size_t)(m0 + ar) * DMODEL + k0 + ac], &As[1 - p][ar][ac]);
            async_ld16(&W[(size_t)(k0 + br0) * N + n0 + bc0],   &Bs[1 - p][br0][bc0]);
            async_ld16(&W[(size_t)(k0 + br1) * N + n0 + bc1],   &Bs[1 - p][br1][bc1]);
        }

        #pragma unroll
        for (int kk = 0; kk < 4; ++kk) {
            // A fragment: lane ln holds row (wm+ln), K pair starting at kk*4+2*hl
            v2f a = *(const v2f*)&As[p][wm + ln][kk * 4 + 2 * hl];
            #pragma unroll
            for (int t = 0; t < 4; ++t) {
                v2f b;
                b.x = Bs[p][kk * 4 + 2 * hl][wn + t * 16 + ln];
                b.y = Bs[p][kk * 4 + 2 * hl + 1][wn + t * 16 + ln];
                acc[t] = WMMA4(a, b, acc[t]);
            }
        }
    }

    // Epilogue: C layout -> lane ln holds col, vgpr i holds row i + 8*hl.
    // Fused RoPE for q/k; scatter into head-major [B,H,S,64].
    #pragma unroll
    for (int t = 0; t < 4; ++t) {
        int n     = n0 + wn + t * 16 + ln;       // global output column
        int which = n / DMODEL;                  // 0=q 1=k 2=v (uniform per tile)
        int c     = n % DMODEL;
        int hh    = c / HDIM;
        int d     = c % HDIM;
        #pragma unroll
        for (int i = 0; i < 8; ++i) {
            int row = m0 + wm + i + 8 * hl;      // global row
            int bb  = row >> 11;                 // /SEQ
            int s   = row & (SEQ - 1);
            float val = acc[t][i];
            float partner = __shfl_xor(val, 1, 32);   // RoPE pair lives in lane^1
            if (which < 2) {
                int   j    = d >> 1;
                float invf = __expf(-(float)j * (LOG10K / 32.0f)); // 10000^(-j/32)
                float ang  = (float)s * invf;
                float sn, cs;
                __sincosf(ang, &sn, &cs);
                val = (d & 1) ? (partner * sn + val * cs)    // out_odd  = e*sin + o*cos
                              : (val * cs - partner * sn);   // out_even = e*cos - o*sin
            }
            float* dst = (which == 0) ? Q : (which == 1) ? K : V;
            dst[(((size_t)bb * NHEAD + hh) * SEQ + s) * HDIM + d] = val;
        }
    }
}

// ===========================================================================
// Kernel 2: flash attention, one wave per (b, h, 16-row q tile).
// S tile = Q(16x64) @ K_blk(16x64)^T via 16 x wmma(16x16x4),
// online softmax with shuffle reductions, P routed C-layout->A-layout via LDS,
// O += P @ V_blk via 16 x wmma.  K/V re-reads hit the 192MB L2.
// ===========================================================================
__global__ __launch_bounds__(32) void attn_kernel(
    const float* __restrict__ Q,
    const float* __restrict__ K,
    const float* __restrict__ V,
    float* __restrict__ Y)        // [8192, 1024] row-major, heads re-interleaved
{
    __shared__ float P[16][16];

    const int lane = threadIdx.x;
    const int hl   = lane >> 4;
    const int ln   = lane & 15;
    const int qt   = blockIdx.x & 127;           // q tile 0..127
    const int bh   = blockIdx.x >> 7;            // 0..63
    const int bb   = bh >> 4, hh = bh & 15;

    const float* Qp = Q + (size_t)bh * SEQ * HDIM;
    const float* Kp = K + (size_t)bh * SEQ * HDIM;
    const float* Vp = V + (size_t)bh * SEQ * HDIM;
    const int q0 = qt * 16;

    // Preload Q fragments with softmax scale folded in (1/sqrt(64) = 0.125)
    v2f qf[16];
    #pragma unroll
    for (int kk = 0; kk < 16; ++kk) {
        v2f v = *(const v2f*)&Qp[(size_t)(q0 + ln) * HDIM + kk * 4 + 2 * hl];
        qf[kk] = v * 0.125f;
    }

    float mrow[8], lrow[8];
    #pragma unroll
    for (int i = 0; i < 8; ++i) { mrow[i] = -1e30f; lrow[i] = 0.0f; }
    v8f acc[4] = {};

    for (int kb = 0; kb <= qt; ++kb) {
        const int kbase = kb * 16;
        if (kb < qt)  // prefetch next K block into cache
            __builtin_prefetch(&Kp[(size_t)(kbase + 16 + ln) * HDIM], 0, 1);

        // ---- scores S = Q @ K^T (scaled) ----
        v8f sc = {};
        #pragma unroll
        for (int kk = 0; kk < 16; ++kk) {
            v2f bfr = *(const v2f*)&Kp[(size_t)(kbase + ln) * HDIM + kk * 4 + 2 * hl];
            sc = WMMA4(qf[kk], bfr, sc);
        }

        // ---- causal mask on diagonal block (col=ln, row=i+8*hl) ----
        if (kb == qt) {
            #pragma unroll
            for (int i = 0; i < 8; ++i)
                if (ln > i + 8 * hl) sc[i] = -1e30f;
        }

        // ---- online softmax (row spans the 16 lanes of a half-wave) ----
        float alpha[8];
        #pragma unroll
        for (int i = 0; i < 8; ++i) {
            float x = sc[i];
            x = fmaxf(x, __shfl_xor(x, 1, 32));
            x = fmaxf(x, __shfl_xor(x, 2, 32));
            x = fmaxf(x, __shfl_xor(x, 4, 32));
            x = fmaxf(x, __shfl_xor(x, 8, 32));
            float mn = fmaxf(mrow[i], x);
            alpha[i] = __expf(mrow[i] - mn);
            mrow[i]  = mn;
            float p  = __expf(sc[i] - mn);
            sc[i]    = p;
            float rs = p;
            rs += __shfl_xor(rs, 1, 32);
            rs += __shfl_xor(rs, 2, 32);
            rs += __shfl_xor(rs, 4, 32);
            rs += __shfl_xor(rs, 8, 32);
            lrow[i] = lrow[i] * alpha[i] + rs;
        }
        #pragma unroll
        for (int t = 0; t < 4; ++t)
            #pragma unroll
            for (int i = 0; i < 8; ++i)
                acc[t][i] *= alpha[i];

        // ---- P: C-layout -> A-layout through LDS ----
        __syncthreads();
        #pragma unroll
        for (int i = 0; i < 8; ++i) P[i + 8 * hl][ln] = sc[i];
        __syncthreads();
        v2f pa[4];
        #pragma unroll
        for (int kk = 0; kk < 4; ++kk)
            pa[kk] = *(const v2f*)&P[ln][kk * 4 + 2 * hl];

        // ---- O += P @ V ----
        #pragma unroll
        for (int t = 0; t < 4; ++t) {
            #pragma unroll
            for (int kk = 0; kk < 4; ++kk) {
                int vr = kbase + kk * 4 + 2 * hl;
                v2f bfr;
                bfr.x = Vp[(size_t)vr * HDIM + t * 16 + ln];
                bfr.y = Vp[(size_t)(vr + 1) * HDIM + t * 16 + ln];
                acc[t] = WMMA4(pa[kk], bfr, acc[t]);
            }
        }
    }

    // ---- normalize and write back re-interleaved to [B*S, 1024] ----
    #pragma unroll
    for (int t = 0; t < 4; ++t)
        #pragma unroll
        for (int i = 0; i < 8; ++i) {
            int row = i + 8 * hl;
            float inv = 1.0f / lrow[i];
            Y[((size_t)bb * SEQ + q0 + row) * DMODEL + hh * HDIM + t * 16 + ln] =
                acc[t][i] * inv;
        }
}

// ===========================================================================
// Kernel 3: out = Y @ W_proj  (same async double-buffered fp32 WMMA GEMM)
// ===========================================================================
__global__ __launch_bounds__(256) void proj_kernel(
    const float* __restrict__ X,   // [8192, 1024]
    const float* __restrict__ W,   // [1024, 1024]
    float* __restrict__ O)         // [8192, 1024]
{
    const int N = DMODEL;
    __shared__ __align__(16) float As[2][64][16];
    __shared__ __align__(16) float Bs[2][16][128];

    const int tid  = threadIdx.x;
    const int lane = tid & 31;
    const int hl   = lane >> 4;
    const int ln   = lane & 15;
    const int wave = tid >> 5;
    const int wm   = (wave & 3) * 16;
    const int wn   = (wave >> 2) * 64;
    const int m0   = blockIdx.x * 64;
    const int n0   = blockIdx.y * 128;

    const int ar = tid >> 2, ac = (tid & 3) * 4;
    const int br0 = tid >> 5,        bc0 = (tid & 31) * 4;
    const int br1 = (tid + 256) >> 5, bc1 = bc0;

    v8f acc[4] = {};

    {
        const int k0 = 0;
        async_ld16(&X[(size_t)(m0 + ar) * DMODEL + k0 + ac], &As[0][ar][ac]);
        async_ld16(&W[(size_t)(k0 + br0) * N + n0 + bc0],   &Bs[0][br0][bc0]);
        async_ld16(&W[(size_t)(k0 + br1) * N + n0 + bc1],   &Bs[0][br1][bc1]);
    }

    for (int kt = 0; kt < DMODEL / 16; ++kt) {
        const int p = kt & 1;
        wait_async0();
        __syncthreads();

        if (kt + 1 < DMODEL / 16) {
            const int k0 = (kt + 1) * 16;
            async_ld16(&X[(size_t)(m0 + ar) * DMODEL + k0 + ac], &As[1 - p][ar][ac]);
            async_ld16(&W[(size_t)(k0 + br0) * N + n0 + bc0],   &Bs[1 - p][br0][bc0]);
            async_ld16(&W[(size_t)(k0 + br1) * N + n0 + bc1],   &Bs[1 - p][br1][bc1]);
        }

        #pragma unroll
        for (int kk = 0; kk < 4; ++kk) {
            v2f a = *(const v2f*)&As[p][wm + ln][kk * 4 + 2 * hl];
            #pragma unroll
            for (int t = 0; t < 4; ++t) {
                v2f b;
                b.x = Bs[p][kk * 4 + 2 * hl][wn + t * 16 + ln];
                b.y = Bs[p][kk * 4 + 2 * hl + 1][wn + t * 16 + ln];
                acc[t] = WMMA4(a, b, acc[t]);
            }
        }
    }

    #pragma unroll
    for (int t = 0; t < 4; ++t) {
        int n = n0 + wn + t * 16 + ln;
        #pragma unroll
        for (int i = 0; i < 8; ++i) {
            int row = m0 + wm + i + 8 * hl;
            O[(size_t)row * N + n] = acc[t][i];
        }
    }
}

// ===========================================================================
extern "C" void kernel_launch(void* const* d_in, const int* in_sizes, int n_in,
                              void* d_out, int out_size, void* d_ws, size_t ws_size,
                              hipStream_t stream)
{
    (void)in_sizes; (void)n_in; (void)out_size; (void)ws_size;

    const float* x     = (const float*)d_in[0];   // [4,2048,1024]
    const float* Wqkv  = (const float*)d_in[1];   // [1024,3072]
    const float* Wproj = (const float*)d_in[2];   // [1024,1024]
    float* out = (float*)d_out;                   // [4,2048,1024]

    const size_t TEN = (size_t)B_SZ * NHEAD * SEQ * HDIM;  // 8388608 floats
    float* Q = (float*)d_ws;
    float* K = Q + TEN;
    float* V = K + TEN;
    float* Y = V + TEN;                                    // total 134 MB of ws

    dim3 blk(256);
    dim3 g1(MROWS / 64, (3 * DMODEL) / 128);   // 128 x 24
    qkv_rope_kernel<<<g1, blk, 0, stream>>>(x, Wqkv, Q, K, V);

    attn_kernel<<<dim3(B_SZ * NHEAD * (SEQ / 16)), dim3(32), 0, stream>>>(Q, K, V, Y);

    dim3 g2(MROWS / 64, DMODEL / 128);         // 128 x 8
    proj_kernel<<<g2, blk, 0, stream>>>(Y, Wproj, out);
}